// DecoderRNN_893353197931
// MI455X (gfx1250) — compile-verified
//
#include <hip/hip_runtime.h>
#include <math.h>

#define HDIM 1024
#define VDIM 128000

typedef __attribute__((ext_vector_type(2))) float v2f;
typedef __attribute__((ext_vector_type(8))) float v8f;

// ---------------------------------------------------------------------------
// x = relu(emb[id]) ; 4 blocks x 256
// ---------------------------------------------------------------------------
__global__ __launch_bounds__(256) void k_embed_relu(const int* __restrict__ ids,
                                                    const float* __restrict__ emb,
                                                    float* __restrict__ x) {
    int i = blockIdx.x * 256 + threadIdx.x;
    int id = ids[0];
    float v = emb[(size_t)id * HDIM + i];
    x[i] = v > 0.f ? v : 0.f;
}

// ---------------------------------------------------------------------------
// out[r] = dot(W[r,:], x) + bias[r]   via V_WMMA_F32_16X16X4_F32, N padded to 16.
// Wave handles 16 rows; block = 8 waves = 128 rows. K = HDIM = 1024.
// A tile (16x4 f32, 2 VGPRs): lanes 0-15 -> K=0,1 ; lanes 16-31 -> K=2,3 (ISA 7.12.2).
// B tile (4x16, 2 VGPRs): only column N=0 nonzero = h[k..k+3].
// D column 0: lane 0 holds M=v (VGPR v), lane 16 holds M=v+8.
// ---------------------------------------------------------------------------
__global__ __launch_bounds__(256) void k_gemv_wmma(const float* __restrict__ W,
                                                   const float* __restrict__ x,
                                                   const float* __restrict__ bias,
                                                   float* __restrict__ out) {
    __shared__ float xs[HDIM];
    const int t = threadIdx.x;
#pragma unroll
    for (int i = 0; i < HDIM / 256; ++i) xs[t + 256 * i] = x[t + 256 * i];
    __syncthreads();

    const int lane = t & 31;
    const int wave = t >> 5;
    const int m    = lane & 15;
    const int kb   = (lane >> 4) << 1;                 // 0 for lanes 0-15, 2 for 16-31
    const int r0   = (blockIdx.x * 8 + wave) * 16;     // first of 16 rows for this wave
    const float* wr = W + (size_t)(r0 + m) * HDIM + kb;
    const bool ncol = (m == 0);                        // this lane carries column N=0

    v8f acc = {0.f, 0.f, 0.f, 0.f, 0.f, 0.f, 0.f, 0.f};
    for (int k = 0; k < HDIM; k += 4) {
        v2f a  = *(const v2f*)(wr + k);                // global b64, row-streaming
        v2f hx = *(const v2f*)(xs + k + kb);           // LDS b64 broadcast per half-wave
        v2f b;
        b.x = ncol ? hx.x : 0.f;                       // select keeps EXEC full for WMMA
        b.y = ncol ? hx.y : 0.f;
        acc = __builtin_amdgcn_wmma_f32_16x16x4_f32(
            /*neg_a=*/false, a, /*neg_b=*/false, b,
            /*c_mod=*/(short)0, acc, /*reuse_a=*/false, /*reuse_b=*/false);
    }

    if (ncol) {                                        // lanes 0 and 16 own column 0
        int base = r0 + ((lane >> 4) << 3);            // lane0 -> rows r0.., lane16 -> r0+8..
#pragma unroll
        for (int v = 0; v < 8; ++v) out[base + v] = acc[v] + bias[base + v];
    }
}

// ---------------------------------------------------------------------------
// GRU gate fusion: gi/gh already include their biases. 4 blocks x 256.
// ---------------------------------------------------------------------------
__global__ __launch_bounds__(256) void k_gru(const float* __restrict__ gi,
                                             const float* __restrict__ gh,
                                             const float* __restrict__ h0,
                                             float* __restrict__ h1) {
    int i = blockIdx.x * 256 + threadIdx.x;
    float r = 1.f / (1.f + expf(-(gi[i] + gh[i])));
    float z = 1.f / (1.f + expf(-(gi[HDIM + i] + gh[HDIM + i])));
    float n = tanhf(gi[2 * HDIM + i] + r * gh[2 * HDIM + i]);
    h1[i] = (1.f - z) * n + z * h0[i];
}

// ---------------------------------------------------------------------------
// log-softmax reductions: 250 blocks x 512 elems, then 1-block finishers.
// ---------------------------------------------------------------------------
__global__ __launch_bounds__(256) void k_pmax(const float* __restrict__ logits,
                                              float* __restrict__ pmax) {
    __shared__ float s[256];
    int t = threadIdx.x;
    int base = blockIdx.x * 512;
    s[t] = fmaxf(logits[base + t], logits[base + 256 + t]);
    __syncthreads();
    for (int o = 128; o > 0; o >>= 1) {
        if (t < o) s[t] = fmaxf(s[t], s[t + o]);
        __syncthreads();
    }
    if (t == 0) pmax[blockIdx.x] = s[0];
}

__global__ __launch_bounds__(256) void k_rmax(const float* __restrict__ pmax,
                                              float* __restrict__ gmax) {
    __shared__ float s[256];
    int t = threadIdx.x;
    s[t] = (t < 250) ? pmax[t] : -INFINITY;
    __syncthreads();
    for (int o = 128; o > 0; o >>= 1) {
        if (t < o) s[t] = fmaxf(s[t], s[t + o]);
        __syncthreads();
    }
    if (t == 0) gmax[0] = s[0];
}

__global__ __launch_bounds__(256) void k_psum(const float* __restrict__ logits,
                                              const float* __restrict__ gmax,
                                              float* __restrict__ psum) {
    __shared__ float s[256];
    int t = threadIdx.x;
    int base = blockIdx.x * 512;
    float gm = gmax[0];
    s[t] = expf(logits[base + t] - gm) + expf(logits[base + 256 + t] - gm);
    __syncthreads();
    for (int o = 128; o > 0; o >>= 1) {
        if (t < o) s[t] += s[t + o];
        __syncthreads();
    }
    if (t == 0) psum[blockIdx.x] = s[0];
}

__global__ __launch_bounds__(256) void k_rsum(const float* __restrict__ psum,
                                              const float* __restrict__ gmax,
                                              float* __restrict__ lse) {
    __shared__ float s[256];
    int t = threadIdx.x;
    s[t] = (t < 250) ? psum[t] : 0.f;
    __syncthreads();
    for (int o = 128; o > 0; o >>= 1) {
        if (t < o) s[t] += s[t + o];
        __syncthreads();
    }
    if (t == 0) lse[0] = gmax[0] + logf(s[0]);
}

__global__ __launch_bounds__(256) void k_logp(float* __restrict__ out,
                                              const float* __restrict__ lse) {
    int i = blockIdx.x * 256 + threadIdx.x;
    out[i] = out[i] - lse[0];   // logits fully rewritten earlier this call -> idempotent per call
}

// ---------------------------------------------------------------------------
extern "C" void kernel_launch(void* const* d_in, const int* in_sizes, int n_in,
                              void* d_out, int out_size, void* d_ws, size_t ws_size,
                              hipStream_t stream) {
    const int*   ids    = (const int*)d_in[0];
    const float* hidden = (const float*)d_in[1];
    const float* emb    = (const float*)d_in[2];
    const float* W_ih   = (const float*)d_in[3];
    const float* W_hh   = (const float*)d_in[4];
    const float* b_ih   = (const float*)d_in[5];
    const float* b_hh   = (const float*)d_in[6];
    const float* W_out  = (const float*)d_in[7];
    const float* b_out  = (const float*)d_in[8];

    float* out = (float*)d_out;           // [0,V): logp ; [V,V+H): h1
    float* ws  = (float*)d_ws;
    float* x    = ws;                     // 1024
    float* gi   = ws + 1024;              // 3072
    float* gh   = ws + 4096;              // 3072
    float* pmax = ws + 7168;              // 256
    float* gmax = ws + 7424;              // 1
    float* psum = ws + 7456;              // 256
    float* lse  = ws + 7712;              // 1   (total ~31 KB of ws)

    float* logits = out;                  // write logits in place, then subtract lse
    float* h1     = out + VDIM;

    k_embed_relu<<<HDIM / 256, 256, 0, stream>>>(ids, emb, x);
    k_gemv_wmma<<<(3 * HDIM) / 128, 256, 0, stream>>>(W_ih, x, b_ih, gi);
    k_gemv_wmma<<<(3 * HDIM) / 128, 256, 0, stream>>>(W_hh, hidden, b_hh, gh);
    k_gru<<<HDIM / 256, 256, 0, stream>>>(gi, gh, hidden, h1);
    k_gemv_wmma<<<VDIM / 128, 256, 0, stream>>>(W_out, h1, b_out, logits);
    k_pmax<<<VDIM / 512, 256, 0, stream>>>(logits, pmax);
    k_rmax<<<1, 256, 0, stream>>>(pmax, gmax);
    k_psum<<<VDIM / 512, 256, 0, stream>>>(logits, gmax, psum);
    k_rsum<<<1, 256, 0, stream>>>(psum, gmax, lse);
    k_logp<<<VDIM / 256, 256, 0, stream>>>(out, lse);
}